// DrugGNN_20426864459786
// MI455X (gfx1250) — compile-verified
//
#include <hip/hip_runtime.h>
#include <hip/hip_bf16.h>

typedef __attribute__((ext_vector_type(2))) float v2f;
typedef __attribute__((ext_vector_type(8))) float v8f;

#define N_NODES 50000
#define N_EDGES 800000
#define HID 64
#define NCLS 32

// ---------------- workspace layout (4-byte units) ----------------
#define OFF_DEG      0          //  int[50176]
#define OFF_ROWSTART 50176      //  int[50176] (uses 50001)
#define OFF_CURSOR   100352     //  int[50176]
#define OFF_RDEG     150528     //  float[50176]
#define OFF_EDGESRC  200704     //  int[800000]
#define OFF_AGG      1000704    //  float[50000*64]  (h_neigh, reused both layers)
#define OFF_H        4200704    //  float[50000*64]
// total ~29.6 MB

// ---------------- zero fill (ints) ----------------
__global__ void zero_int_kernel(int* __restrict__ p, int n) {
    int i = blockIdx.x * blockDim.x + threadIdx.x;
    int stride = gridDim.x * blockDim.x;
    for (; i < n; i += stride) p[i] = 0;
}

// ---------------- in-degree histogram (integer atomics) ----------------
__global__ void count_kernel(const int* __restrict__ dst, int* __restrict__ deg, int n_edges) {
    int e = blockIdx.x * blockDim.x + threadIdx.x;
    if (e < n_edges) atomicAdd(&deg[dst[e]], 1);
}

// ---------------- single-workgroup exclusive scan -> row_start, cursor, rdeg ----------------
__global__ void scan_kernel(const int* __restrict__ deg,
                            int* __restrict__ row_start,
                            int* __restrict__ cursor,
                            float* __restrict__ rdeg,
                            int n) {
    __shared__ int sums[256];
    __shared__ int pre[257];
    const int t = threadIdx.x;
    const int CH = (n + 255) / 256;
    const int beg = t * CH;
    const int end = (beg + CH < n) ? beg + CH : n;

    int s = 0;
    for (int i = beg; i < end; ++i) s += deg[i];
    sums[t] = s;
    __syncthreads();
    if (t == 0) {
        int acc = 0;
        for (int i = 0; i < 256; ++i) { pre[i] = acc; acc += sums[i]; }
        pre[256] = acc;
    }
    __syncthreads();
    int acc = pre[t];
    for (int i = beg; i < end; ++i) {
        row_start[i] = acc;
        cursor[i]    = acc;
        int d = deg[i];
        rdeg[i] = 1.0f / fmaxf((float)d, 1.0f);
        acc += d;
    }
    if (t == 0) row_start[n] = pre[256];
}

// ---------------- CSR bucket fill (integer atomics) ----------------
__global__ void fill_kernel(const int* __restrict__ src, const int* __restrict__ dst,
                            int* __restrict__ cursor, int* __restrict__ edge_src,
                            int n_edges) {
    int e = blockIdx.x * blockDim.x + threadIdx.x;
    if (e < n_edges) {
        int pos = atomicAdd(&cursor[dst[e]], 1);
        edge_src[pos] = src[e];
    }
}

// ---------------- mean aggregation by gather: one wave per node ----------------
// hneigh[n][f] = rdeg[n] * sum_{e in CSR[n]} X[edge_src[e]][f]     (64 feats, 2/lane)
__global__ void gather_mean_kernel(const float* __restrict__ X,        // [N,64]
                                   const int* __restrict__ row_start,  // [N+1]
                                   const int* __restrict__ edge_src,   // [E]
                                   const float* __restrict__ rdeg,     // [N]
                                   float* __restrict__ hneigh,         // [N,64]
                                   int n_nodes) {
    const int wave = blockIdx.x * (blockDim.x >> 5) + (threadIdx.x >> 5);
    if (wave >= n_nodes) return;
    const int lane = threadIdx.x & 31;
    const int beg = row_start[wave];
    const int end = row_start[wave + 1];

    v2f sum = (v2f)0.0f;
    int i = beg;
    // 2-way unroll for a bit of ILP on the L2-resident row gathers
    for (; i + 2 <= end; i += 2) {
        int s0 = edge_src[i];
        int s1 = edge_src[i + 1];
        v2f a = *(const v2f*)(X + (long)s0 * 64 + lane * 2);
        v2f b = *(const v2f*)(X + (long)s1 * 64 + lane * 2);
        sum += a;
        sum += b;
    }
    if (i < end) {
        int s0 = edge_src[i];
        sum += *(const v2f*)(X + (long)s0 * 64 + lane * 2);
    }
    const float rd = rdeg[wave];
    sum.x *= rd;
    sum.y *= rd;
    *(v2f*)(hneigh + (long)wave * 64 + lane * 2) = sum;
}

// ---------------- fused SAGE dense layer via f32 WMMA ----------------
// Y[16 rows x NOUT] per wave:  Y = X@Wself + hneigh@Wneigh + b  (ReLU optional)
// A fragment (16x4 f32, 2 VGPRs): lane L<16 -> (M=L, K=kk..kk+1); lane>=16 -> (M=L-16, K=kk+2..kk+3)
// B fragment (4x16 f32, 2 VGPRs): mirrored, N striped over lanes.
// C/D (16x16 f32, 8 VGPRs): vgpr r = row r (lanes 0-15) / row r+8 (lanes 16-31).
template <int NOUT, bool RELU>
__global__ void sage_layer_kernel(const float* __restrict__ X,       // [N,64]
                                  const float* __restrict__ hneigh,  // [N,64] pre-scaled
                                  const float* __restrict__ Wself,   // [64,NOUT] row-major
                                  const float* __restrict__ Wneigh,  // [64,NOUT]
                                  const float* __restrict__ bias,    // [NOUT]
                                  float* __restrict__ Y,             // [N,NOUT]
                                  int n_nodes) {
    constexpr int NT = NOUT / 16;
    const int lane = threadIdx.x & 31;
    const int wave = blockIdx.x * (blockDim.x >> 5) + (threadIdx.x >> 5);
    const int base = wave * 16;
    if (base >= n_nodes) return;   // uniform per wave -> EXEC stays all-1s for WMMA

    const int l15  = lane & 15;
    const int half = lane >> 4;
    const int row  = base + l15;

    const float* __restrict__ xr = X      + (long)row * 64;
    const float* __restrict__ ar = hneigh + (long)row * 64;

    v8f acc[NT];
#pragma unroll
    for (int t = 0; t < NT; ++t) acc[t] = (v8f)0.0f;

#pragma unroll
    for (int k = 0; k < 16; ++k) {
        const int kk = k * 4 + half * 2;
        v2f a  = *(const v2f*)(xr + kk);
        v2f hn = *(const v2f*)(ar + kk);
#pragma unroll
        for (int t = 0; t < NT; ++t) {
            const int col = t * 16 + l15;
            v2f bs;
            bs.x = Wself[(long)kk * NOUT + col];
            bs.y = Wself[(long)(kk + 1) * NOUT + col];
            acc[t] = __builtin_amdgcn_wmma_f32_16x16x4_f32(
                false, a, false, bs, (short)0, acc[t], false, false);
            v2f bn;
            bn.x = Wneigh[(long)kk * NOUT + col];
            bn.y = Wneigh[(long)(kk + 1) * NOUT + col];
            acc[t] = __builtin_amdgcn_wmma_f32_16x16x4_f32(
                false, hn, false, bn, (short)0, acc[t], false, false);
        }
    }

#pragma unroll
    for (int t = 0; t < NT; ++t) {
        const int col = t * 16 + l15;
        const float bv = bias[col];
#pragma unroll
        for (int r = 0; r < 8; ++r) {
            const int m = r + half * 8;
            float v = acc[t][r] + bv;
            if (RELU) v = fmaxf(v, 0.0f);
            Y[(long)(base + m) * NOUT + col] = v;
        }
    }
}

extern "C" void kernel_launch(void* const* d_in, const int* in_sizes, int n_in,
                              void* d_out, int out_size, void* d_ws, size_t ws_size,
                              hipStream_t stream) {
    const float* features = (const float*)d_in[0];   // [50000,64]
    const float* W_self1  = (const float*)d_in[1];   // [64,64]
    const float* W_neigh1 = (const float*)d_in[2];   // [64,64]
    const float* b1       = (const float*)d_in[3];   // [64]
    const float* W_self2  = (const float*)d_in[4];   // [64,32]
    const float* W_neigh2 = (const float*)d_in[5];   // [64,32]
    const float* b2       = (const float*)d_in[6];   // [32]
    const int*   src      = (const int*)d_in[7];     // [800000]
    const int*   dst      = (const int*)d_in[8];     // [800000]
    float*       out      = (float*)d_out;           // [50000,32]

    int*   deg       = (int*)d_ws + OFF_DEG;
    int*   row_start = (int*)d_ws + OFF_ROWSTART;
    int*   cursor    = (int*)d_ws + OFF_CURSOR;
    float* rdeg      = (float*)d_ws + OFF_RDEG;
    int*   edge_src  = (int*)d_ws + OFF_EDGESRC;
    float* agg       = (float*)d_ws + OFF_AGG;
    float* h         = (float*)d_ws + OFF_H;

    const int edge_blocks = (N_EDGES + 255) / 256;
    const int node_wave_blocks = (N_NODES + 7) / 8;          // 8 waves (256 thr) per block
    const int tiles = (N_NODES + 15) / 16;
    const int layer_blocks = (tiles + 3) / 4;                // 4 waves (128 thr) per block

    // ---- build CSR (per call; integer atomics only) ----
    zero_int_kernel<<<64, 256, 0, stream>>>(deg, N_NODES);
    count_kernel<<<edge_blocks, 256, 0, stream>>>(dst, deg, N_EDGES);
    scan_kernel<<<1, 256, 0, stream>>>(deg, row_start, cursor, rdeg, N_NODES);
    fill_kernel<<<edge_blocks, 256, 0, stream>>>(src, dst, cursor, edge_src, N_EDGES);

    // ---- layer 1 ----
    gather_mean_kernel<<<node_wave_blocks, 256, 0, stream>>>(
        features, row_start, edge_src, rdeg, agg, N_NODES);
    sage_layer_kernel<HID, true><<<layer_blocks, 128, 0, stream>>>(
        features, agg, W_self1, W_neigh1, b1, h, N_NODES);

    // ---- layer 2 ----
    gather_mean_kernel<<<node_wave_blocks, 256, 0, stream>>>(
        h, row_start, edge_src, rdeg, agg, N_NODES);
    sage_layer_kernel<NCLS, false><<<layer_blocks, 128, 0, stream>>>(
        h, agg, W_self2, W_neigh2, b2, out, N_NODES);
}